// E2PNKPConv_80470507258245
// MI455X (gfx1250) — compile-verified
//
#include <hip/hip_runtime.h>
#include <hip/hip_bf16.h>

// KPConv encoder-decoder forward for MI455X (gfx1250).
// - Heavy GEMMs (agg x W, Wsc shortcuts, decoder unary blocks) run on the
//   matrix pipe via V_WMMA_F32_16X16X4_F32 (fp32-exact, per reference dtype),
//   register-tiled 2x2 (32x32 C per wave) for A/B fragment reuse.
// - Gather/influence/aggregate, group-norm+residual+leaky are VALU kernels
//   with LDS staging.

#define SLOPE 0.1f
#define MAXK 40

typedef float v2f __attribute__((ext_vector_type(2)));
typedef float v8f __attribute__((ext_vector_type(8)));

// ---------------------------------------------------------------------------
// Block 0 special case: Cin == 1, Mc = 15 (not WMMA friendly). Fully fused.
// One 64-thread block per query point.
// ---------------------------------------------------------------------------
__global__ void kpconv_c1_kernel(const float* __restrict__ qp,
                                 const float* __restrict__ sp,
                                 const float* __restrict__ feats,
                                 const int* __restrict__ nbr, int K,
                                 const float* __restrict__ kp, float invSigma,
                                 const float* __restrict__ W,   // [15,64]
                                 float* __restrict__ out)       // [Nq,64]
{
    __shared__ float sP[MAXK * 3];
    __shared__ float sF[MAXK];
    __shared__ float sAgg[15];
    const int n = blockIdx.x;
    const int t = threadIdx.x;
    if (t < K) {
        int j = nbr[(size_t)n * K + t];
        sP[t * 3 + 0] = sp[(size_t)j * 3 + 0];
        sP[t * 3 + 1] = sp[(size_t)j * 3 + 1];
        sP[t * 3 + 2] = sp[(size_t)j * 3 + 2];
        sF[t] = feats[j];
    }
    __syncthreads();
    const float qx = qp[(size_t)n * 3 + 0];
    const float qy = qp[(size_t)n * 3 + 1];
    const float qz = qp[(size_t)n * 3 + 2];
    if (t < 15) {
        const float kx = kp[t * 3 + 0], ky = kp[t * 3 + 1], kz = kp[t * 3 + 2];
        float acc = 0.f;
        for (int k = 0; k < K; ++k) {
            float dx = sP[k * 3 + 0] - qx - kx;
            float dy = sP[k * 3 + 1] - qy - ky;
            float dz = sP[k * 3 + 2] - qz - kz;
            float d = sqrtf(fmaf(dx, dx, fmaf(dy, dy, dz * dz)) + 1e-12f);
            float infl = fmaxf(0.f, 1.f - d * invSigma);
            acc = fmaf(infl, sF[k], acc);
        }
        sAgg[t] = acc;
    }
    __syncthreads();
    float o = 0.f;
#pragma unroll
    for (int m = 0; m < 15; ++m) o = fmaf(sAgg[m], W[m * 64 + t], o);
    out[(size_t)n * 64 + t] = o;
}

// ---------------------------------------------------------------------------
// KPConv aggregation: agg[nloc, m*Cin + c] = sum_k infl(n,k,m) * feats[nbr,c]
// One block per query point (chunked by row0), grid.y tiles channels by 128.
// ---------------------------------------------------------------------------
__global__ void kpconv_agg_kernel(const float* __restrict__ qp,
                                  const float* __restrict__ sp,
                                  const float* __restrict__ sf,
                                  const int* __restrict__ nbr, int K,
                                  const float* __restrict__ kp, float invSigma,
                                  int Cin, int row0,
                                  float* __restrict__ agg)
{
    __shared__ int   sN[MAXK];
    __shared__ float sP[MAXK * 3];
    __shared__ float sI[MAXK * 15];
    const int nloc = blockIdx.x;
    const int n = row0 + nloc;
    const int t = threadIdx.x;
    if (t < K) {
        int j = nbr[(size_t)n * K + t];
        sN[t] = j;
        sP[t * 3 + 0] = sp[(size_t)j * 3 + 0];
        sP[t * 3 + 1] = sp[(size_t)j * 3 + 1];
        sP[t * 3 + 2] = sp[(size_t)j * 3 + 2];
    }
    __syncthreads();
    const float qx = qp[(size_t)n * 3 + 0];
    const float qy = qp[(size_t)n * 3 + 1];
    const float qz = qp[(size_t)n * 3 + 2];
    for (int idx = t; idx < K * 15; idx += blockDim.x) {
        int k = idx / 15;
        int m = idx - k * 15;
        float dx = sP[k * 3 + 0] - qx - kp[m * 3 + 0];
        float dy = sP[k * 3 + 1] - qy - kp[m * 3 + 1];
        float dz = sP[k * 3 + 2] - qz - kp[m * 3 + 2];
        float d = sqrtf(fmaf(dx, dx, fmaf(dy, dy, dz * dz)) + 1e-12f);
        sI[idx] = fmaxf(0.f, 1.f - d * invSigma);
    }
    __syncthreads();
    const int c = blockIdx.y * 128 + t;
    if (c >= Cin) return;
    float acc[15];
#pragma unroll
    for (int m = 0; m < 15; ++m) acc[m] = 0.f;
    for (int k = 0; k < K; ++k) {
        float f = sf[(size_t)sN[k] * Cin + c];
#pragma unroll
        for (int m = 0; m < 15; ++m) acc[m] = fmaf(sI[k * 15 + m], f, acc[m]);
    }
    size_t base = (size_t)nloc * (size_t)(15 * Cin) + c;
#pragma unroll
    for (int m = 0; m < 15; ++m) agg[base + (size_t)m * Cin] = acc[m];
}

// ---------------------------------------------------------------------------
// fp32 GEMM on the WMMA pipe: C[nrows,ncols] = A[nrows,Kdim] * B[Kdim,ncols]
// V_WMMA_F32_16X16X4_F32, register-tiled: each wave computes a 32x32 C patch
// (2x2 grid of 16x16 tiles, 4 v8f accumulators) so each A/B fragment feeds
// two WMMAs. 4 waves/block -> 128 output columns per block.
// Requirements (all call sites satisfy): Kdim % 16 == 0, ncols % 128 == 0.
// Fragment layouts per CDNA5 ISA 7.12.2:
//   A 16x4 : lanes 0-15 row M, vgpr{0,1} = K{0,1}; lanes 16-31 = K{2,3}
//   B 4x16 : lanes 0-15 col N for K{0,1}; lanes 16-31 for K{2,3}
//   C 16x16: vgpr r -> row r (lanes 0-15) / row r+8 (lanes 16-31)
// ---------------------------------------------------------------------------
__global__ void gemm_wmma_f32_kernel(const float* __restrict__ A, int lda,
                                     const float* __restrict__ B, int ldb,
                                     float* __restrict__ C, int ldc,
                                     int nrows, int Kdim,
                                     const float* __restrict__ bias)
{
    const int lane = threadIdx.x & 31;
    const int wave = threadIdx.x >> 5;
    const int row0 = blockIdx.x * 32;
    const int col0 = (blockIdx.y * 4 + wave) * 32;
    const int lh = lane & 15;
    const int hi = lane >> 4;  // 0: K pair {0,1}, 1: K pair {2,3}

    // clamp A row indices; padded rows are discarded at store time
    int ar0 = row0 + lh;
    int ar1 = row0 + 16 + lh;
    if (ar0 > nrows - 1) ar0 = nrows - 1;
    if (ar1 > nrows - 1) ar1 = nrows - 1;
    const float* Ap0 = A + (size_t)ar0 * lda + 2 * hi;
    const float* Ap1 = A + (size_t)ar1 * lda + 2 * hi;
    const float* Bp0 = B + (size_t)(2 * hi) * ldb + col0 + lh;
    const float* Bp1 = Bp0 + 16;

    v8f acc00 = {0.f, 0.f, 0.f, 0.f, 0.f, 0.f, 0.f, 0.f};
    v8f acc01 = {0.f, 0.f, 0.f, 0.f, 0.f, 0.f, 0.f, 0.f};
    v8f acc10 = {0.f, 0.f, 0.f, 0.f, 0.f, 0.f, 0.f, 0.f};
    v8f acc11 = {0.f, 0.f, 0.f, 0.f, 0.f, 0.f, 0.f, 0.f};

    for (int k0 = 0; k0 < Kdim; k0 += 16) {
        if (k0 + 16 < Kdim) {  // near-cache prefetch of next K chunk
            __builtin_prefetch(Ap0 + k0 + 16, 0, 3);
            __builtin_prefetch(Ap1 + k0 + 16, 0, 3);
            __builtin_prefetch(Bp0 + (size_t)(k0 + 16) * ldb, 0, 3);
        }
#pragma unroll
        for (int kk = 0; kk < 16; kk += 4) {
            v2f a0, a1, b0, b1;
            a0.x = Ap0[k0 + kk];
            a0.y = Ap0[k0 + kk + 1];
            a1.x = Ap1[k0 + kk];
            a1.y = Ap1[k0 + kk + 1];
            const size_t bo = (size_t)(k0 + kk) * ldb;
            b0.x = Bp0[bo];
            b0.y = Bp0[bo + ldb];
            b1.x = Bp1[bo];
            b1.y = Bp1[bo + ldb];
            acc00 = __builtin_amdgcn_wmma_f32_16x16x4_f32(false, a0, false, b0,
                                                          (short)0, acc00, false, false);
            acc01 = __builtin_amdgcn_wmma_f32_16x16x4_f32(false, a0, false, b1,
                                                          (short)0, acc01, false, false);
            acc10 = __builtin_amdgcn_wmma_f32_16x16x4_f32(false, a1, false, b0,
                                                          (short)0, acc10, false, false);
            acc11 = __builtin_amdgcn_wmma_f32_16x16x4_f32(false, a1, false, b1,
                                                          (short)0, acc11, false, false);
        }
    }

    const float bv0 = bias ? bias[col0 + lh] : 0.f;
    const float bv1 = bias ? bias[col0 + 16 + lh] : 0.f;
#pragma unroll
    for (int r = 0; r < 8; ++r) {
        const int rA = row0 + r + hi * 8;
        const int rB = rA + 16;
        if (rA < nrows) {
            C[(size_t)rA * ldc + col0 + lh] = acc00[r] + bv0;
            C[(size_t)rA * ldc + col0 + 16 + lh] = acc01[r] + bv1;
        }
        if (rB < nrows) {
            C[(size_t)rB * ldc + col0 + lh] = acc10[r] + bv0;
            C[(size_t)rB * ldc + col0 + 16 + lh] = acc11[r] + bv1;
        }
    }
}

// ---------------------------------------------------------------------------
// GroupNorm(32) + optional residual + LeakyReLU(0.1), in place. One block/row.
// sc == nullptr  -> no residual.  scIdx == nullptr -> identity row index.
// ---------------------------------------------------------------------------
__global__ void gnorm_leaky_kernel(float* __restrict__ x, int C,
                                   const float* __restrict__ g,
                                   const float* __restrict__ b,
                                   const float* __restrict__ sc,
                                   const int* __restrict__ scIdx, int scStride)
{
    __shared__ float rowv[1024];
    __shared__ float mu[32];
    __shared__ float rs[32];
    const int n = blockIdx.x;
    const int t = threadIdx.x;
    for (int c = t; c < C; c += blockDim.x) rowv[c] = x[(size_t)n * C + c];
    __syncthreads();
    const int gs = C >> 5;  // channels per group (2..32, power of two)
    if (t < 32) {
        float s = 0.f, s2 = 0.f;
        for (int j = 0; j < gs; ++j) {
            float v = rowv[t * gs + j];
            s += v;
            s2 = fmaf(v, v, s2);
        }
        float m = s / (float)gs;
        mu[t] = m;
        rs[t] = rsqrtf(s2 / (float)gs - m * m + 1e-5f);
    }
    __syncthreads();
    int srow = n;
    if (scIdx) srow = scIdx[(size_t)n * scStride];
    for (int c = t; c < C; c += blockDim.x) {
        int gi = c / gs;
        float v = fmaf((rowv[c] - mu[gi]) * rs[gi], g[c], b[c]);
        if (sc) v += sc[(size_t)srow * C + c];
        x[(size_t)n * C + c] = (v > 0.f) ? v : SLOPE * v;
    }
}

// ---------------------------------------------------------------------------
// out[n] = concat(upf[up[n]], skip[n])   (nearest upsample + concat)
// ---------------------------------------------------------------------------
__global__ void concat_up_kernel(const float* __restrict__ upf,
                                 const int* __restrict__ up,
                                 const float* __restrict__ skip,
                                 float* __restrict__ out, int C1, int C2)
{
    const int n = blockIdx.x;
    const int j = up[n];
    const int CT = C1 + C2;
    for (int c = threadIdx.x; c < CT; c += blockDim.x)
        out[(size_t)n * CT + c] =
            (c < C1) ? upf[(size_t)j * C1 + c] : skip[(size_t)n * C2 + (c - C1)];
}

// ---------------------------------------------------------------------------
extern "C" void kernel_launch(void* const* d_in, const int* in_sizes, int n_in,
                              void* d_out, int out_size, void* d_ws, size_t ws_size,
                              hipStream_t stream)
{
    (void)out_size; (void)ws_size;
    const int N1 = 20000, N2 = 5000, N3 = 1250, N4 = 320;

    const float* feats = (const float*)d_in[0];
    const float* pts1 = (const float*)d_in[1];
    const float* pts2 = (const float*)d_in[2];
    const float* pts3 = (const float*)d_in[3];
    const float* pts4 = (const float*)d_in[4];
    const int* nbr1 = (const int*)d_in[5];
    const int* nbr2 = (const int*)d_in[6];
    const int* nbr3 = (const int*)d_in[7];
    const int* nbr4 = (const int*)d_in[8];
    const int* sub2 = (const int*)d_in[9];
    const int* sub3 = (const int*)d_in[10];
    const int* sub4 = (const int*)d_in[11];
    const int* up2 = (const int*)d_in[12];
    const int* up3 = (const int*)d_in[13];

    static const int   DOUT[11] = {64,128,128,256,256,256,512,512,512,1024,1024};
    static const float SIGV[11] = {0.75f,0.75f,0.75f,1.5f,1.5f,1.5f,3.f,3.f,3.f,6.f,6.f};
    static const int   HASWSC[11] = {0,1,0,1,0,0,1,0,0,1,0};

    // params pytree flattening: detect sorted-key (jax tree_flatten) vs
    // insertion-order leaves via the first params leaf size
    // (blocks[0]['W'] = 15*1*64 = 960 vs blocks[0]['kp'] = 45).
    const float *W[11], *Wsc[11], *gg[11], *bb[11], *kp[11];
    int p = 14;
    const bool sortedKeys = (n_in > 14 && in_sizes[14] == 960);
    for (int i = 0; i < 11; ++i) {
        if (sortedKeys) {  // W, (Wsc), b, g, kp
            W[i] = (const float*)d_in[p++];
            Wsc[i] = HASWSC[i] ? (const float*)d_in[p++] : nullptr;
            bb[i] = (const float*)d_in[p++];
            gg[i] = (const float*)d_in[p++];
            kp[i] = (const float*)d_in[p++];
        } else {           // kp, W, g, b, (Wsc)
            kp[i] = (const float*)d_in[p++];
            W[i] = (const float*)d_in[p++];
            gg[i] = (const float*)d_in[p++];
            bb[i] = (const float*)d_in[p++];
            Wsc[i] = HASWSC[i] ? (const float*)d_in[p++] : nullptr;
        }
    }
    const float *d2W, *d2b, *d3W, *d3b, *d3g;
    if (sortedKeys) {  // dec2{W,b}, dec3{W,b,g}
        d2W = (const float*)d_in[p++]; d2b = (const float*)d_in[p++];
        d3W = (const float*)d_in[p++]; d3b = (const float*)d_in[p++];
        d3g = (const float*)d_in[p++];
    } else {           // dec3{W,g,b}, dec2{W,b}
        d3W = (const float*)d_in[p++]; d3g = (const float*)d_in[p++];
        d3b = (const float*)d_in[p++];
        d2W = (const float*)d_in[p++]; d2b = (const float*)d_in[p++];
    }

    // ---- workspace layout (floats) ----
    float* ws = (float*)d_ws;
    size_t off = 0;
    auto wsAlloc = [&](size_t nElems) { float* q = ws + off; off += nElems; return q; };
    float* f1a = wsAlloc((size_t)N1 * 64);
    float* f1  = wsAlloc((size_t)N1 * 128);
    float* f2a = wsAlloc((size_t)N2 * 128);
    float* f2b = wsAlloc((size_t)N2 * 256);
    float* f2  = wsAlloc((size_t)N2 * 256);
    float* f3a = wsAlloc((size_t)N3 * 256);
    float* f3b = wsAlloc((size_t)N3 * 512);
    float* f3  = wsAlloc((size_t)N3 * 512);
    float* f4a = wsAlloc((size_t)N4 * 512);
    float* f4b = wsAlloc((size_t)N4 * 1024);
    float* scW = wsAlloc((size_t)N1 * 128);          // Wsc shortcut scratch (max size)
    float* cat3 = wsAlloc((size_t)N3 * 1536);
    float* cat2 = wsAlloc((size_t)N2 * 768);
    const size_t AGG_CAP = (size_t)6 * 1024 * 1024;  // 24 MB agg ring
    float* aggB = wsAlloc(AGG_CAP);

    // ---- output regions: (l2, l3, f4) flat in return order ----
    float* l2o = (float*)d_out;                 // [N2,256]
    float* l3o = l2o + (size_t)N2 * 256;        // [N3,512]
    float* f4o = l3o + (size_t)N3 * 512;        // [N4,1024]

    auto gemm = [&](const float* A, int lda, const float* B, int ldb, float* Cc,
                    int ldc, int nrows, int Kd, int ncols, const float* bias) {
        dim3 g((nrows + 31) / 32, ncols / 128);
        gemm_wmma_f32_kernel<<<g, 128, 0, stream>>>(A, lda, B, ldb, Cc, ldc, nrows,
                                                    Kd, bias);
    };
    auto convBlock = [&](const float* qp, int Nq, const float* sp, const float* sf,
                         int Cin, const int* nb, int K, int i, float* outb) {
        const int Cout = DOUT[i];
        const int Mc = 15 * Cin;
        int rowsPer = (int)(AGG_CAP / (size_t)Mc) & ~31;
        if (rowsPer < 32) rowsPer = 32;
        const float invSig = 1.0f / SIGV[i];
        for (int r0 = 0; r0 < Nq; r0 += rowsPer) {
            const int nr = (Nq - r0 < rowsPer) ? (Nq - r0) : rowsPer;
            dim3 gA(nr, (Cin + 127) / 128);
            kpconv_agg_kernel<<<gA, 128, 0, stream>>>(qp, sp, sf, nb, K, kp[i],
                                                      invSig, Cin, r0, aggB);
            gemm(aggB, Mc, W[i], Cout, outb + (size_t)r0 * Cout, Cout, nr, Mc, Cout,
                 nullptr);
        }
    };
    auto gnorm = [&](float* x, int Nq, int C, const float* gv, const float* bv,
                     const float* sc, const int* scIdx, int scStride) {
        gnorm_leaky_kernel<<<Nq, 256, 0, stream>>>(x, C, gv, bv, sc, scIdx, scStride);
    };

    // ---- encoder ----
    // block0: 1 -> 64, no residual
    kpconv_c1_kernel<<<N1, 64, 0, stream>>>(pts1, pts1, feats, nbr1, 26, kp[0],
                                            1.0f / SIGV[0], W[0], f1a);
    gnorm(f1a, N1, 64, gg[0], bb[0], nullptr, nullptr, 0);
    // block1: 64 -> 128, residual via Wsc
    gemm(f1a, 64, Wsc[1], 128, scW, 128, N1, 64, 128, nullptr);
    convBlock(pts1, N1, pts1, f1a, 64, nbr1, 26, 1, f1);
    gnorm(f1, N1, 128, gg[1], bb[1], scW, nullptr, 0);
    // block2: pooled (sub2), 128 -> 128, residual = f1[sub2[:,0]]
    convBlock(pts2, N2, pts1, f1, 128, sub2, 26, 2, f2a);
    gnorm(f2a, N2, 128, gg[2], bb[2], f1, sub2, 26);
    // block3: 128 -> 256, residual via Wsc
    gemm(f2a, 128, Wsc[3], 256, scW, 256, N2, 128, 256, nullptr);
    convBlock(pts2, N2, pts2, f2a, 128, nbr2, 31, 3, f2b);
    gnorm(f2b, N2, 256, gg[3], bb[3], scW, nullptr, 0);
    // block4: 256 -> 256, identity residual
    convBlock(pts2, N2, pts2, f2b, 256, nbr2, 31, 4, f2);
    gnorm(f2, N2, 256, gg[4], bb[4], f2b, nullptr, 0);
    // block5: pooled (sub3), 256 -> 256
    convBlock(pts3, N3, pts2, f2, 256, sub3, 31, 5, f3a);
    gnorm(f3a, N3, 256, gg[5], bb[5], f2, sub3, 31);
    // block6: 256 -> 512, Wsc
    gemm(f3a, 256, Wsc[6], 512, scW, 512, N3, 256, 512, nullptr);
    convBlock(pts3, N3, pts3, f3a, 256, nbr3, 33, 6, f3b);
    gnorm(f3b, N3, 512, gg[6], bb[6], scW, nullptr, 0);
    // block7: 512 -> 512, identity residual
    convBlock(pts3, N3, pts3, f3b, 512, nbr3, 33, 7, f3);
    gnorm(f3, N3, 512, gg[7], bb[7], f3b, nullptr, 0);
    // block8: pooled (sub4), 512 -> 512
    convBlock(pts4, N4, pts3, f3, 512, sub4, 33, 8, f4a);
    gnorm(f4a, N4, 512, gg[8], bb[8], f3, sub4, 33);
    // block9: 512 -> 1024, Wsc
    gemm(f4a, 512, Wsc[9], 1024, scW, 1024, N4, 512, 1024, nullptr);
    convBlock(pts4, N4, pts4, f4a, 512, nbr4, 37, 9, f4b);
    gnorm(f4b, N4, 1024, gg[9], bb[9], scW, nullptr, 0);
    // block10: 1024 -> 1024, identity residual -> f4 output region
    convBlock(pts4, N4, pts4, f4b, 1024, nbr4, 37, 10, f4o);
    gnorm(f4o, N4, 1024, gg[10], bb[10], f4b, nullptr, 0);

    // ---- decoder ----
    // l3 = leaky(gnorm(concat(f4[up3], f3) @ d3W))
    concat_up_kernel<<<N3, 256, 0, stream>>>(f4o, up3, f3, cat3, 1024, 512);
    gemm(cat3, 1536, d3W, 512, l3o, 512, N3, 1536, 512, nullptr);
    gnorm(l3o, N3, 512, d3g, d3b, nullptr, nullptr, 0);
    // l2 = concat(l3[up2], f2) @ d2W + d2b   (no norm / activation)
    concat_up_kernel<<<N2, 256, 0, stream>>>(l3o, up2, f2, cat2, 512, 256);
    gemm(cat2, 768, d2W, 256, l2o, 256, N2, 768, 256, d2b);
}